// ConditionedMessagePassingLayer_34299608825888
// MI455X (gfx1250) — compile-verified
//
#include <hip/hip_runtime.h>
#include <hip/hip_bf16.h>

#define NNODE 40000
#define NEDGE 640000
#define INV_SQRT3 0.5773502691896258f
#define INV_SQRT2 0.7071067811865476f
#define INV_AVG   0.0625f   // 1/16

typedef float v2f __attribute__((ext_vector_type(2)));
typedef float v8f __attribute__((ext_vector_type(8)));

__device__ __forceinline__ v8f wmma4(v2f a, v2f b, v8f c) {
    // D = A(16x4,f32) * B(4x16,f32) + C(16x16,f32)
    return __builtin_amdgcn_wmma_f32_16x16x4_f32(false, a, false, b, (short)0, c,
                                                 false, false);
}

// Fast transcendental helpers: v_exp_f32 + v_rcp_f32, no precise-divide chain.
__device__ __forceinline__ float fast_rcp(float x)  { return __builtin_amdgcn_rcpf(x); }
__device__ __forceinline__ float silu(float x)      { return x * fast_rcp(1.0f + __expf(-x)); }
__device__ __forceinline__ float fast_tanh(float x) { return 2.0f * fast_rcp(1.0f + __expf(-2.0f * x)) - 1.0f; }

// ---------------------------------------------------------------- zero scratch
__global__ __launch_bounds__(256) void zero_kernel(float* __restrict__ p, int n) {
    int i = blockIdx.x * 256 + threadIdx.x;
    if (i < n) p[i] = 0.0f;
}

// ---------------------------------------------------------------- edge kernel
// block = 256 threads = 8 waves, each wave handles 16 edges.
// LDS: edge_w1 (512 f) | edge_w2 (10240 f) | per-wave scratch 2560 f
__global__ __launch_bounds__(256) void edge_kernel(
    const float* __restrict__ nf,     // [N,128]
    const float* __restrict__ eattr,  // [E,4]
    const float* __restrict__ elen,   // [E,8]
    const int*   __restrict__ eidx,   // [2,E]
    const float* __restrict__ w1g,    // [8,64]
    const float* __restrict__ w2g,    // [64,160]
    float* __restrict__ p0,           // [N,64]
    float* __restrict__ p1)           // [N,96,3]
{
    extern __shared__ float sm[];
    float* w1s  = sm;                                   // 512
    float* w2s  = sm + 512;                             // 10240
    float* wbuf = sm + 512 + 10240 + (threadIdx.x >> 5) * 2560;

    const int lane = threadIdx.x & 31;
    const int wave = threadIdx.x >> 5;
    const int m  = lane & 15;
    const int kh = lane >> 4;

    const int e0 = (blockIdx.x * 8 + wave) * 16;

    // Prefetch DTP-phase gather lines (global_prefetch_b8) so the random
    // node_features rows are in-flight while the MLP WMMAs execute.
    {
        const int e_  = lane >> 1;
        const int hf_ = lane & 1;
        const int eg_ = e0 + e_;
        __builtin_prefetch(eattr + eg_ * 4, 0, 3);
        const int snd_ = eidx[NEDGE + eg_];
        __builtin_prefetch(nf + (size_t)snd_ * 128 + hf_ * 32, 0, 3);
        __builtin_prefetch(nf + (size_t)snd_ * 128 + 64 + hf_ * 32, 0, 3);
    }

    for (int i = threadIdx.x; i < 512;   i += 256) w1s[i] = w1g[i];
    for (int i = threadIdx.x; i < 10240; i += 256) w2s[i] = w2g[i];
    __syncthreads();

    // ---- GEMM1: elen[16x8] @ w1[8x64]  (2 k-steps, 4 n-tiles)
    v8f h[4] = {};
    #pragma unroll
    for (int ks = 0; ks < 2; ++ks) {
        const int kb = ks * 4 + kh * 2;
        v2f a; a.x = elen[(e0 + m) * 8 + kb];
               a.y = elen[(e0 + m) * 8 + kb + 1];
        #pragma unroll
        for (int t = 0; t < 4; ++t) {
            v2f b; b.x = w1s[kb * 64 + t * 16 + m];
                   b.y = w1s[(kb + 1) * 64 + t * 16 + m];
            h[t] = wmma4(a, b, h[t]);
        }
    }
    // silu + relayout C->A via LDS (first 1024 floats of wbuf)
    #pragma unroll
    for (int t = 0; t < 4; ++t)
        #pragma unroll
        for (int r = 0; r < 8; ++r)
            wbuf[(r + 8 * kh) * 64 + t * 16 + m] = silu(h[t][r]);
    __syncthreads();

    // ---- GEMM2: h[16x64] @ w2[64x160]  (16 k-steps, 10 n-tiles)
    v8f c[10] = {};
    #pragma unroll
    for (int ks = 0; ks < 16; ++ks) {
        const int kb = ks * 4 + kh * 2;
        v2f a; a.x = wbuf[m * 64 + kb];
               a.y = wbuf[m * 64 + kb + 1];
        #pragma unroll
        for (int t = 0; t < 10; ++t) {
            v2f b; b.x = w2s[kb * 160 + t * 16 + m];
                   b.y = w2s[(kb + 1) * 160 + t * 16 + m];
            c[t] = wmma4(a, b, c[t]);
        }
    }
    __syncthreads();
    #pragma unroll
    for (int t = 0; t < 10; ++t)
        #pragma unroll
        for (int r = 0; r < 8; ++r)
            wbuf[(r + 8 * kh) * 160 + t * 16 + m] = c[t][r];
    __syncthreads();

    // ---- DTP + atomic scatter: 2 lanes per edge, 16 channels each
    const int e  = lane >> 1;
    const int hf = lane & 1;
    const int eg = e0 + e;
    const int recv = eidx[eg];
    const int snd  = eidx[NEDGE + eg];
    const float y0  = eattr[eg * 4 + 0];
    const float y1x = eattr[eg * 4 + 1];
    const float y1y = eattr[eg * 4 + 2];
    const float y1z = eattr[eg * 4 + 3];
    const float* wrow = wbuf + e * 160;
    const float* nfr  = nf + (size_t)snd * 128;
    float* p0r = p0 + (size_t)recv * 64;
    float* p1r = p1 + (size_t)recv * 288;

    #pragma unroll 4
    for (int u = 0; u < 16; ++u) {
        const int uu = hf * 16 + u;
        const float su = nfr[uu];
        const float vx = nfr[32 + uu * 3 + 0];
        const float vy = nfr[32 + uu * 3 + 1];
        const float vz = nfr[32 + uu * 3 + 2];
        const float wv1 = wrow[uu],       wv2 = wrow[32 + uu];
        const float wv3 = wrow[64 + uu],  wv4 = wrow[96 + uu];
        const float wv5 = wrow[128 + uu];

        unsafeAtomicAdd(&p0r[uu], wv1 * su * y0 * INV_AVG);
        const float dv = vx * y1x + vy * y1y + vz * y1z;
        unsafeAtomicAdd(&p0r[32 + uu], wv4 * dv * INV_SQRT3 * INV_AVG);

        const float a2 = wv2 * su * INV_AVG;
        unsafeAtomicAdd(&p1r[uu * 3 + 0], a2 * y1x);
        unsafeAtomicAdd(&p1r[uu * 3 + 1], a2 * y1y);
        unsafeAtomicAdd(&p1r[uu * 3 + 2], a2 * y1z);

        const float a3 = wv3 * y0 * INV_AVG;
        unsafeAtomicAdd(&p1r[(32 + uu) * 3 + 0], a3 * vx);
        unsafeAtomicAdd(&p1r[(32 + uu) * 3 + 1], a3 * vy);
        unsafeAtomicAdd(&p1r[(32 + uu) * 3 + 2], a3 * vz);

        const float cx = vy * y1z - vz * y1y;
        const float cy = vz * y1x - vx * y1z;
        const float cz = vx * y1y - vy * y1x;
        const float a5 = wv5 * INV_SQRT2 * INV_AVG;
        unsafeAtomicAdd(&p1r[(64 + uu) * 3 + 0], a5 * cx);
        unsafeAtomicAdd(&p1r[(64 + uu) * 3 + 1], a5 * cy);
        unsafeAtomicAdd(&p1r[(64 + uu) * 3 + 2], a5 * cz);
    }
}

// ---------------------------------------------------------------- node kernel
// block = 1 wave (32 threads), handles 16 nodes.
// LDS floats: hbuf 1024 | wv 6656 | q0 2560 | q1 12288 | u 2560  => 25088
__global__ __launch_bounds__(32) void node_kernel(
    const float* __restrict__ p0,      // [N,64]
    const float* __restrict__ p1,      // [N,96,3]
    const float* __restrict__ velEmb,  // [N,4]
    const float* __restrict__ velLen,  // [N,8]
    const float* __restrict__ vw1,     // [8,64]
    const float* __restrict__ vw2,     // [64,416]
    const float* __restrict__ lw0,     // [160,64]
    const float* __restrict__ lw1,     // [256,32]
    float* __restrict__ out)           // [N,128]
{
    extern __shared__ float sm[];
    float* hbuf = sm;               // 1024
    float* wv   = sm + 1024;        // 6656  : w_vel [16][416]
    float* q0b  = wv + 6656;        // 2560  : q0 [16][160]
    float* q1b  = q0b + 2560;       // 12288 : q1 [3][16][256]
    float* ub   = q1b + 12288;      // 2560  : u0 [16][64] + u1 [16][32][3]

    const int lane = threadIdx.x;
    const int m  = lane & 15;
    const int kh = lane >> 4;
    const int n0 = blockIdx.x * 16;

    // Prefetch DTP-phase rows while the MLP WMMAs run.
    {
        const int nd_ = lane >> 1;
        const int hf_ = lane & 1;
        const int gn_ = n0 + nd_;
        __builtin_prefetch(velEmb + gn_ * 4, 0, 3);
        __builtin_prefetch(p0 + (size_t)gn_ * 64 + hf_ * 32, 0, 3);
        __builtin_prefetch(p1 + (size_t)gn_ * 288 + hf_ * 144, 0, 3);
        __builtin_prefetch(p1 + (size_t)gn_ * 288 + hf_ * 144 + 72, 0, 3);
    }

    // ---- GEMM1: velLen[16x8] @ vw1[8x64]
    v8f h[4] = {};
    #pragma unroll
    for (int ks = 0; ks < 2; ++ks) {
        const int kb = ks * 4 + kh * 2;
        v2f a; a.x = velLen[(n0 + m) * 8 + kb];
               a.y = velLen[(n0 + m) * 8 + kb + 1];
        #pragma unroll
        for (int t = 0; t < 4; ++t) {
            v2f b; b.x = vw1[kb * 64 + t * 16 + m];
                   b.y = vw1[(kb + 1) * 64 + t * 16 + m];
            h[t] = wmma4(a, b, h[t]);
        }
    }
    #pragma unroll
    for (int t = 0; t < 4; ++t)
        #pragma unroll
        for (int r = 0; r < 8; ++r)
            hbuf[(r + 8 * kh) * 64 + t * 16 + m] = silu(h[t][r]);
    __syncthreads();

    // ---- GEMM2: h[16x64] @ vw2[64x416], 26 n-tiles in 2 groups of 13
    for (int g = 0; g < 2; ++g) {
        v8f c[13] = {};
        #pragma unroll
        for (int ks = 0; ks < 16; ++ks) {
            const int kb = ks * 4 + kh * 2;
            v2f a; a.x = hbuf[m * 64 + kb];
                   a.y = hbuf[m * 64 + kb + 1];
            #pragma unroll
            for (int t = 0; t < 13; ++t) {
                const int col = (g * 13 + t) * 16 + m;
                v2f b; b.x = vw2[kb * 416 + col];
                       b.y = vw2[(kb + 1) * 416 + col];
                c[t] = wmma4(a, b, c[t]);
            }
        }
        #pragma unroll
        for (int t = 0; t < 13; ++t)
            #pragma unroll
            for (int r = 0; r < 8; ++r)
                wv[(r + 8 * kh) * 416 + (g * 13 + t) * 16 + m] = c[t][r];
    }
    __syncthreads();

    // ---- node DTP: 2 lanes per node
    {
        const int nd = lane >> 1;
        const int hf = lane & 1;
        const int gn = n0 + nd;
        const float z0  = velEmb[gn * 4 + 0];
        const float z1x = velEmb[gn * 4 + 1];
        const float z1y = velEmb[gn * 4 + 2];
        const float z1z = velEmb[gn * 4 + 3];
        const float* p0r  = p0 + (size_t)gn * 64;
        const float* p1r  = p1 + (size_t)gn * 288;
        const float* wrow = wv + nd * 416;
        float* q0r = q0b + nd * 160;
        float* q1x = q1b + 0 * 4096 + nd * 256;
        float* q1y = q1b + 1 * 4096 + nd * 256;
        float* q1z = q1b + 2 * 4096 + nd * 256;

        for (int j = 0; j < 32; ++j) {                 // scalar channels
            const int u = hf * 32 + j;
            const float s  = p0r[u];
            const float a1 = wrow[u];                  // w1
            const float a2 = wrow[64 + u];             // w2
            q0r[u]  = a1 * s * z0;
            q1x[u]  = a2 * s * z1x;
            q1y[u]  = a2 * s * z1y;
            q1z[u]  = a2 * s * z1z;
        }
        for (int j = 0; j < 48; ++j) {                 // vector channels
            const int u = hf * 48 + j;
            const float vx = p1r[u * 3 + 0];
            const float vy = p1r[u * 3 + 1];
            const float vz = p1r[u * 3 + 2];
            const float a3 = wrow[128 + u];            // w3
            const float a4 = wrow[224 + u];            // w4
            const float a5 = wrow[320 + u];            // w5
            const float dv = vx * z1x + vy * z1y + vz * z1z;
            q0r[64 + u]  = a4 * dv * INV_SQRT3;
            q1x[64 + u]  = a3 * vx * z0;
            q1y[64 + u]  = a3 * vy * z0;
            q1z[64 + u]  = a3 * vz * z0;
            const float cx = vy * z1z - vz * z1y;
            const float cy = vz * z1x - vx * z1z;
            const float cz = vx * z1y - vy * z1x;
            q1x[160 + u] = a5 * cx * INV_SQRT2;
            q1y[160 + u] = a5 * cy * INV_SQRT2;
            q1z[160 + u] = a5 * cz * INV_SQRT2;
        }
    }
    __syncthreads();

    // ---- u0 = q0[16x160] @ lw0[160x64]  (40 k-steps, 4 n-tiles)
    {
        v8f c[4] = {};
        #pragma unroll 8
        for (int ks = 0; ks < 40; ++ks) {
            const int kb = ks * 4 + kh * 2;
            v2f a; a.x = q0b[m * 160 + kb];
                   a.y = q0b[m * 160 + kb + 1];
            #pragma unroll
            for (int t = 0; t < 4; ++t) {
                v2f b; b.x = lw0[kb * 64 + t * 16 + m];
                       b.y = lw0[(kb + 1) * 64 + t * 16 + m];
                c[t] = wmma4(a, b, c[t]);
            }
        }
        #pragma unroll
        for (int t = 0; t < 4; ++t)
            #pragma unroll
            for (int r = 0; r < 8; ++r)
                ub[(r + 8 * kh) * 64 + t * 16 + m] = c[t][r];
    }

    // ---- u1: per channel, q1[16x256] @ lw1[256x32]  (64 k-steps, 2 n-tiles)
    for (int ch = 0; ch < 3; ++ch) {
        v8f c[2] = {};
        #pragma unroll 8
        for (int ks = 0; ks < 64; ++ks) {
            const int kb = ks * 4 + kh * 2;
            v2f a; a.x = q1b[ch * 4096 + m * 256 + kb];
                   a.y = q1b[ch * 4096 + m * 256 + kb + 1];
            #pragma unroll
            for (int t = 0; t < 2; ++t) {
                v2f b; b.x = lw1[kb * 32 + t * 16 + m];
                       b.y = lw1[(kb + 1) * 32 + t * 16 + m];
                c[t] = wmma4(a, b, c[t]);
            }
        }
        #pragma unroll
        for (int t = 0; t < 2; ++t)
            #pragma unroll
            for (int r = 0; r < 8; ++r)
                ub[1024 + (r + 8 * kh) * 96 + (t * 16 + m) * 3 + ch] = c[t][r];
    }
    __syncthreads();

    // ---- gating + output: 16 nodes x 128 features
    for (int i = 0; i < 64; ++i) {
        const int idx  = i * 32 + lane;
        const int node = idx >> 7;
        const int f    = idx & 127;
        float o;
        if (f < 32) {
            o = fast_tanh(ub[node * 64 + f]);
        } else {
            const int j = f - 32;
            const int v = j / 3, ch = j % 3;
            o = fast_tanh(ub[node * 64 + 32 + v]) * ub[1024 + node * 96 + v * 3 + ch];
        }
        out[(size_t)(n0 + node) * 128 + f] = o;
    }
}

// ---------------------------------------------------------------- launch
extern "C" void kernel_launch(void* const* d_in, const int* in_sizes, int n_in,
                              void* d_out, int out_size, void* d_ws, size_t ws_size,
                              hipStream_t stream) {
    const float* nf     = (const float*)d_in[0];
    const float* eattr  = (const float*)d_in[1];
    const float* elen   = (const float*)d_in[2];
    const float* velEmb = (const float*)d_in[3];
    const float* velLen = (const float*)d_in[4];
    const int*   eidx   = (const int*)  d_in[5];
    const float* ew1    = (const float*)d_in[6];
    const float* ew2    = (const float*)d_in[7];
    const float* vw1    = (const float*)d_in[8];
    const float* vw2    = (const float*)d_in[9];
    const float* lw0    = (const float*)d_in[10];
    const float* lw1    = (const float*)d_in[11];
    float* out = (float*)d_out;

    float* p0 = (float*)d_ws;                       // [N,64]
    float* p1 = p0 + (size_t)NNODE * 64;            // [N,96,3]

    zero_kernel<<<55000, 256, 0, stream>>>(p0, NNODE * 352);

    const size_t edge_smem = (512 + 10240 + 8 * 2560) * sizeof(float);   // 124928 B
    edge_kernel<<<NEDGE / 128, 256, edge_smem, stream>>>(
        nf, eattr, elen, eidx, ew1, ew2, p0, p1);

    const size_t node_smem = 25088 * sizeof(float);                      // 100352 B
    node_kernel<<<NNODE / 16, 32, node_smem, stream>>>(
        p0, p1, velEmb, velLen, vw1, vw2, lw0, lw1, out);
}